// LinearTextEmbedding_57604101374655
// MI455X (gfx1250) — compile-verified
//
#include <hip/hip_runtime.h>
#include <stdint.h>

// Output: (48, 1024, 1024) f32. out[flat] = |bits[flat % 4096]| > 0.5 ? 1 : 0.
// Pure store-bandwidth problem: 201 MB of writes, ~16 KB of reads.
// Strategy: each thread owns one float4 of the 1024-float4 periodic pattern
// (its grid stride is a multiple of 1024 float4s, so the slot is loop
// invariant), stages its 16 B of `bits` via CDNA5 async global->LDS copy,
// thresholds in registers once, then emits only non-temporal b128 stores.

typedef float v4f __attribute__((ext_vector_type(4)));

#define TABLE_V4 1024   // 4096 bits / 4 per float4
#define BLOCK    256
#define GRID     2048   // total threads = 524288 -> multiple of TABLE_V4

__global__ __launch_bounds__(BLOCK)
void bitmap_broadcast_kernel(const float* __restrict__ bits,
                             float* __restrict__ out,
                             long nvec4)
{
    __shared__ float lds_bits[BLOCK * 4];   // 4 KB staging: one float4 per thread

    const int  tid   = threadIdx.x;
    const long gid   = (long)blockIdx.x * BLOCK + tid;
    const long total = (long)GRID * BLOCK;          // multiple of TABLE_V4

    // This thread's slot in the periodic pattern (invariant across iterations).
    const int tslot = (int)(gid & (TABLE_V4 - 1));

    // ---- CDNA5 async copy: global -> LDS, tracked by ASYNCcnt ----
    // Low 32 bits of a generic pointer into __shared__ are the LDS byte offset.
    unsigned lds_off = (unsigned)(uintptr_t)(&lds_bits[tid * 4]);
    uint64_t gaddr   = (uint64_t)(uintptr_t)(bits + (long)tslot * 4);
    asm volatile(
        "global_load_async_to_lds_b128 %0, %1, off"
        :
        : "v"(lds_off), "v"(gaddr)
        : "memory");
    asm volatile("s_wait_asynccnt 0" ::: "memory");

    // Read back own 16 B from LDS (ds_load_b128), threshold in registers.
    v4f b = *(const v4f*)(&lds_bits[tid * 4]);
    v4f r;
    r.x = (__builtin_fabsf(b.x) > 0.5f) ? 1.0f : 0.0f;
    r.y = (__builtin_fabsf(b.y) > 0.5f) ? 1.0f : 0.0f;
    r.z = (__builtin_fabsf(b.z) > 0.5f) ? 1.0f : 0.0f;
    r.w = (__builtin_fabsf(b.w) > 0.5f) ? 1.0f : 0.0f;

    // ---- Steady state: pure non-temporal 128-bit stores ----
    // NT: output (192 MiB) is exactly L2-sized and never re-read on device.
    v4f* o = (v4f*)out;
    #pragma unroll 4
    for (long p = gid; p < nvec4; p += total) {
        __builtin_nontemporal_store(r, o + p);
    }
}

extern "C" void kernel_launch(void* const* d_in, const int* in_sizes, int n_in,
                              void* d_out, int out_size, void* d_ws, size_t ws_size,
                              hipStream_t stream) {
    (void)in_sizes; (void)n_in; (void)d_ws; (void)ws_size;
    const float* bits = (const float*)d_in[0];
    float* out = (float*)d_out;
    long nvec4 = (long)out_size / 4;   // 48*1024*1024/4 = 12,582,912 float4 stores
    bitmap_broadcast_kernel<<<GRID, BLOCK, 0, stream>>>(bits, out, nvec4);
}